// RoiPoolingConv_87806311400258
// MI455X (gfx1250) — compile-verified
//
#include <hip/hip_runtime.h>

typedef __attribute__((ext_vector_type(2))) float v2f;
typedef __attribute__((ext_vector_type(8))) float v8f;

#define PH 14
#define PW 14
#define FH 128
#define FW 128
#define FC 512
#define NROI 1000

// One block per (roi, px). 256 threads = 8 waves; each wave computes 4
// channel-tiles of 16. Per tile: D[16x16] (py x ch) = Wy[16x32] @ T[32x16]
// via 8x V_WMMA_F32_16X16X4_F32, where T is the x-interpolated crop column
// and Wy holds the bilinear y-weights (zero for padded rows automatically).
__global__ __launch_bounds__(256) void roi_pool_wmma(
    const float* __restrict__ fmap,   // [128,128,512]
    const int*   __restrict__ rois,   // [1000,4] (x1,y1,x2,y2)
    float*       __restrict__ out)    // [1000,14,14,512]
{
  const int roi  = blockIdx.x;
  const int px   = blockIdx.y;
  const int lane = threadIdx.x & 31;
  const int wave = threadIdx.x >> 5;   // 0..7
  const int m    = lane & 15;          // A row (py) and B col (channel-in-tile)
  const int half = lane >> 4;          // K sub-pair selector per ISA layout

  // ROI parameters (uniform per block; SCALE=1.0 so int->float->int is exact)
  const int x0   = rois[roi * 4 + 0];
  const int y0   = rois[roi * 4 + 1];
  const int in_w = rois[roi * 4 + 2] - x0 + 1;   // 2..32
  const int in_h = rois[roi * 4 + 3] - y0 + 1;   // 2..32

  // ---- x interpolation (uniform: px is uniform per block) ----
  const float sx = (float)px * ((float)in_w / (float)PW);
  const int   xt = (int)floorf(sx);
  const float wx = sx - (float)xt;
  int xr = xt + 1; if (xr > in_w - 1) xr = in_w - 1;
  const int cl = x0 + xt;
  const int cr = x0 + xr;

  // ---- y interpolation for this lane's output row m ----
  const float sy = (float)m * ((float)in_h / (float)PH);
  const int   yt = (int)floorf(sy);
  const float wy = sy - (float)yt;
  int yb = yt + 1; if (yb > in_h - 1) yb = in_h - 1;

  // Precompute A-operand (Wy) per K-step: a.x = Wy[m, 4k+2h], a.y = Wy[m, 4k+2h+1].
  // Rows r >= in_h get zero weight automatically (yt,yb < in_h for m < 14).
  v2f aw[8];
#pragma unroll
  for (int ks = 0; ks < 8; ++ks) {
    const int r0 = 4 * ks + 2 * half;
    const int r1 = r0 + 1;
    aw[ks].x = ((r0 == yt) ? (1.0f - wy) : 0.0f) + ((r0 == yb) ? wy : 0.0f);
    aw[ks].y = ((r1 == yt) ? (1.0f - wy) : 0.0f) + ((r1 == yb) ? wy : 0.0f);
  }

  // Source rows this lane reads per K-step: y0 + 4k + 2*half (+1).
  // y0 <= 95 and r <= 31 so y0+r <= 126 < 128: always in-bounds, no masking.
  const int rowbase = (y0 + 2 * half) * FW;               // in columns
  const size_t rstep = (size_t)(4 * FW) * FC;             // 4 rows of floats
  const size_t rnext = (size_t)FW * FC;                   // 1 row of floats

#pragma unroll
  for (int t = 0; t < 4; ++t) {
    const int c0 = (wave + 8 * t) * 16;                   // channel tile base
    const int c  = c0 + m;                                // this lane's channel
    const float* pl = fmap + (size_t)(rowbase + cl) * FC + c;
    const float* pr = fmap + (size_t)(rowbase + cr) * FC + c;

    v8f acc = {0.f, 0.f, 0.f, 0.f, 0.f, 0.f, 0.f, 0.f};
#pragma unroll
    for (int ks = 0; ks < 8; ++ks) {
      const size_t o0 = (size_t)ks * rstep;
      const size_t o1 = o0 + rnext;
      // B-operand: x-interpolated fmap values T[r, c] (fmap is L2-resident: 32MB << 192MB L2)
      const float vl0 = pl[o0], vr0 = pr[o0];
      const float vl1 = pl[o1], vr1 = pr[o1];
      v2f b;
      b.x = vl0 + wx * (vr0 - vl0);
      b.y = vl1 + wx * (vr1 - vl1);
      // D = A*B + C   (f32 WMMA, RNE, full precision)
      acc = __builtin_amdgcn_wmma_f32_16x16x4_f32(
          /*neg_a=*/false, aw[ks], /*neg_b=*/false, b,
          /*c_mod=*/(short)0, acc, /*reuse_a=*/false, /*reuse_b=*/false);
    }

    // Store: VGPR v holds py=v (lanes 0-15) / py=v+8 (lanes 16-31).
    // Non-temporal: 401MB write-once stream must not evict the L2-resident fmap.
#pragma unroll
    for (int v = 0; v < 8; ++v) {
      const int py = v + 8 * half;
      if (py < PH) {
        float* o = out + (((size_t)roi * PH + py) * PW + px) * FC + c;
        __builtin_nontemporal_store(acc[v], o);
      }
    }
  }
}

extern "C" void kernel_launch(void* const* d_in, const int* in_sizes, int n_in,
                              void* d_out, int out_size, void* d_ws, size_t ws_size,
                              hipStream_t stream) {
  const float* img  = (const float*)d_in[0];   // [1,128,128,512] f32
  const int*   rois = (const int*)d_in[1];     // [1000,4] i32
  float*       out  = (float*)d_out;           // [1000,14,14,512] f32
  (void)in_sizes; (void)n_in; (void)out_size; (void)d_ws; (void)ws_size;

  dim3 grid(NROI, PW, 1);
  roi_pool_wmma<<<grid, 256, 0, stream>>>(img, rois, out);
}